// GAT_61959198212462
// MI455X (gfx1250) — compile-verified
//
#include <hip/hip_runtime.h>
#include <hip/hip_bf16.h>
#include <math.h>

// ---------------------------------------------------------------------------
// GAT (3x GATConv + BN + pool + FC) for MI455X / gfx1250, wave32.
// GEMMs use exact-fp32 WMMA (V_WMMA_F32_16X16X4_F32), one wave per 16x64
// output strip (4 accumulators, A-frag reused 4x). Edge phase uses L2 atomics
// (the whole working set fits in the 192MB L2).
// ---------------------------------------------------------------------------

typedef __attribute__((ext_vector_type(2))) float v2f;
typedef __attribute__((ext_vector_type(8))) float v8f;

#define D_HID 256   // H * C = 4 * 64
#define HEADS 4
#define CH    64
#define LRELU_SLOPE 0.2f

// monotonic float <-> uint mapping (for atomicMax on floats incl. negatives)
__device__ __forceinline__ unsigned fflip(float f) {
    unsigned u = __float_as_uint(f);
    return (u & 0x80000000u) ? ~u : (u | 0x80000000u);
}
__device__ __forceinline__ float funflip(unsigned u) {
    u = (u & 0x80000000u) ? (u & 0x7FFFFFFFu) : ~u;
    return __uint_as_float(u);
}
#define NEG_INF_FLIPPED 0x007FFFFFu   // fflip(-inf)

__device__ __forceinline__ v8f wmma_f32(v2f a, v2f b, v8f c) {
    return __builtin_amdgcn_wmma_f32_16x16x4_f32(
        /*neg_a=*/false, a, /*neg_b=*/false, b,
        /*c_mod=*/(short)0, c, /*reuse_a=*/false, /*reuse_b=*/false);
}

// ---------------------------------------------------------------------------
// GEMM: C[M,Ncols] = A[M,K] * B[K,Ncols], fp32 WMMA 16x16x4.
// One wave -> 16(M) x 64(N) strip: A-frag loaded once per k-step (v2f, 8B
// aligned since ka is even), reused by 4 independent WMMAs.
// A-frag (16x4): lanes 0-15 -> M=lane, K={k0,k0+1}; lanes 16-31 -> K={k0+2,k0+3}
// B-frag (4x16): mirrored (lane -> N column, vgpr -> K row)
// C/D (16x16):   vgpr v -> M = v (lanes 0-15) / v+8 (lanes 16-31), N = lane&15
// Rows are clamped (not predicated) so the inner loop is branch-free; clamped
// duplicate rows are never stored.
// ---------------------------------------------------------------------------
__global__ void gemm_wmma_f32(const float* __restrict__ A,
                              const float* __restrict__ B,
                              float* __restrict__ C,
                              int M, int K, int Ncols)
{
    const int wavesPerBlock = blockDim.x >> 5;
    const int waveId = blockIdx.x * wavesPerBlock + (threadIdx.x >> 5);
    const int lane   = threadIdx.x & 31;
    const int stripsN = Ncols >> 6;          // 64-wide strips (4 for 256)
    const int tilesM  = (M + 15) >> 4;
    if (waveId >= tilesM * stripsN) return;
    const int tileM = waveId / stripsN;
    const int strip = waveId % stripsN;

    const int half = lane >> 4;              // 0: lanes 0-15, 1: lanes 16-31
    const int l    = lane & 15;
    int arow = tileM * 16 + l;
    if (arow >= M) arow = M - 1;             // clamp: keep loads in-bounds
    const int bcol = strip * 64 + l;

    const float* __restrict__ Arow = A + (size_t)arow * K + half * 2;
    const float* __restrict__ Bp   = B + bcol;

    v8f acc0 = {}, acc1 = {}, acc2 = {}, acc3 = {};
#pragma unroll 2
    for (int k0 = 0; k0 < K; k0 += 4) {
        const v2f a = *(const v2f*)(Arow + k0);
        const int ka = k0 + half * 2;
        const float* __restrict__ b0p = Bp + ka * Ncols;          // row ka
        const float* __restrict__ b1p = b0p + Ncols;              // row ka+1
        v2f b0, b1, b2, b3;
        b0.x = b0p[0];  b0.y = b1p[0];
        b1.x = b0p[16]; b1.y = b1p[16];
        b2.x = b0p[32]; b2.y = b1p[32];
        b3.x = b0p[48]; b3.y = b1p[48];
        acc0 = wmma_f32(a, b0, acc0);
        acc1 = wmma_f32(a, b1, acc1);
        acc2 = wmma_f32(a, b2, acc2);
        acc3 = wmma_f32(a, b3, acc3);
    }

#pragma unroll
    for (int v = 0; v < 8; ++v) {
        const int row = tileM * 16 + half * 8 + v;
        if (row < M) {
            float* __restrict__ Crow = C + (size_t)row * Ncols + strip * 64 + l;
            Crow[0]  = acc0[v];
            Crow[16] = acc1[v];
            Crow[32] = acc2[v];
            Crow[48] = acc3[v];
        }
    }
}

// attention coefficients: als[n,h] = <h[n,h,:], a_s[h,:]>, same for ald
__global__ void attn_coef(const float* __restrict__ h,
                          const float* __restrict__ aS,
                          const float* __restrict__ aD,
                          float* __restrict__ als, float* __restrict__ ald,
                          int Nn)
{
    const int i = blockIdx.x * blockDim.x + threadIdx.x;  // node*H + head
    if (i >= Nn * HEADS) return;
    const int node = i >> 2, head = i & 3;
    const float* hp = h + (size_t)node * D_HID + head * CH;
    const float* s  = aS + head * CH;
    const float* d  = aD + head * CH;
    float vs = 0.f, vd = 0.f;
#pragma unroll 8
    for (int c = 0; c < CH; ++c) { const float v = hp[c]; vs += v * s[c]; vd += v * d[c]; }
    als[i] = vs;
    ald[i] = vd;
}

// zero/seed the per-layer scratch (agg, seg-max, denom, BN column sums)
__global__ void init_layer(float* __restrict__ agg, unsigned* __restrict__ m,
                           float* __restrict__ denom,
                           float* __restrict__ colsum, float* __restrict__ colsq,
                           int aggN, int Nn)
{
    const int i = blockIdx.x * blockDim.x + threadIdx.x;
    if (i < aggN) agg[i] = 0.f;
    if (i < Nn * HEADS) { m[i] = NEG_INF_FLIPPED; denom[i] = 0.f; }
    if (i < D_HID) { colsum[i] = 0.f; colsq[i] = 0.f; }
}

// pass 1: e = leaky_relu(als[src]+ald[dst]); segment max via atomicMax(flip)
__global__ void edge_logits(const int* __restrict__ src, const int* __restrict__ dst,
                            const float* __restrict__ als, const float* __restrict__ ald,
                            float* __restrict__ e, unsigned* __restrict__ m, int Etot)
{
    const int i = blockIdx.x * blockDim.x + threadIdx.x;   // edge*H + head
    if (i >= Etot * HEADS) return;
    const int eid = i >> 2, head = i & 3;
    const int s = src[eid], d = dst[eid];
    float v = als[s * HEADS + head] + ald[d * HEADS + head];
    v = (v > 0.f) ? v : LRELU_SLOPE * v;
    e[i] = v;
    atomicMax(&m[d * HEADS + head], fflip(v));
}

// pass 2: e = exp(e - max); denom[dst] += e
__global__ void edge_exp(const int* __restrict__ dst, float* __restrict__ e,
                         const unsigned* __restrict__ m, float* __restrict__ denom,
                         int Etot)
{
    const int i = blockIdx.x * blockDim.x + threadIdx.x;
    if (i >= Etot * HEADS) return;
    const int eid = i >> 2, head = i & 3;
    const int d = dst[eid];
    const float mv = funflip(m[d * HEADS + head]);
    const float ex = expf(e[i] - mv);
    e[i] = ex;
    atomicAdd(&denom[d * HEADS + head], ex);
}

// pass 3: agg[dst,h,:] += h[src,h,:] * (e / (denom[dst,h] + 1e-16))
// one wave per (edge, head): 64 channels -> 2 atomics per lane
__global__ void edge_aggregate(const int* __restrict__ src, const int* __restrict__ dst,
                               const float* __restrict__ h, const float* __restrict__ e,
                               const float* __restrict__ denom, float* __restrict__ agg,
                               int Etot)
{
    const int wid  = (blockIdx.x * blockDim.x + threadIdx.x) >> 5;  // edge*H + head
    const int lane = threadIdx.x & 31;
    if (wid >= Etot * HEADS) return;
    const int eid = wid >> 2, head = wid & 3;
    const int s = src[eid], d = dst[eid];
    const float alpha = e[wid] / (denom[d * HEADS + head] + 1e-16f);
    const size_t bs = (size_t)s * D_HID + head * CH;
    const size_t bd = (size_t)d * D_HID + head * CH;
    atomicAdd(&agg[bd + lane],      h[bs + lane]      * alpha);
    atomicAdd(&agg[bd + lane + 32], h[bs + lane + 32] * alpha);
}

// out = relu(agg + bias)
__global__ void post_bias_relu(const float* __restrict__ agg, const float* __restrict__ bias,
                               float* __restrict__ out, int total)
{
    const int i = blockIdx.x * blockDim.x + threadIdx.x;
    if (i >= total) return;
    const float v = agg[i] + bias[i & (D_HID - 1)];
    out[i] = v > 0.f ? v : 0.f;
}

// per-column sum / sum-of-squares (one column per thread, strided over nodes)
__global__ void col_stats(const float* __restrict__ x, float* __restrict__ colsum,
                          float* __restrict__ colsq, int Nn)
{
    const int col = threadIdx.x;              // blockDim.x == 256
    float s = 0.f, sq = 0.f;
    for (int n = blockIdx.x; n < Nn; n += gridDim.x) {
        const float v = x[(size_t)n * D_HID + col];
        s += v; sq += v * v;
    }
    atomicAdd(&colsum[col], s);
    atomicAdd(&colsq[col], sq);
}

// in-place batchnorm
__global__ void bn_apply(float* __restrict__ x, const float* __restrict__ colsum,
                         const float* __restrict__ colsq, const float* __restrict__ g,
                         const float* __restrict__ be, int Nn)
{
    const int i = blockIdx.x * blockDim.x + threadIdx.x;
    if (i >= Nn * D_HID) return;
    const int col = i & (D_HID - 1);
    const float inv = 1.f / (float)Nn;
    const float mu  = colsum[col] * inv;
    const float var = colsq[col] * inv - mu * mu;
    x[i] = g[col] * (x[i] - mu) * rsqrtf(var + 1e-5f) + be[col];
}

__global__ void pool_init(unsigned* __restrict__ pooled, int total)
{
    const int i = blockIdx.x * blockDim.x + threadIdx.x;
    if (i < total) pooled[i] = NEG_INF_FLIPPED;
}

__global__ void pool_max(const float* __restrict__ x, const int* __restrict__ batch,
                         unsigned* __restrict__ pooled, int Nn)
{
    const int i = blockIdx.x * blockDim.x + threadIdx.x;
    if (i >= Nn * D_HID) return;
    const int node = i >> 8, col = i & (D_HID - 1);
    atomicMax(&pooled[batch[node] * D_HID + col], fflip(x[i]));
}

__global__ void fc_sigmoid(const unsigned* __restrict__ pooled,
                           const float* __restrict__ fcw, const float* __restrict__ fcb,
                           float* __restrict__ out, int G)
{
    const int g = blockIdx.x * blockDim.x + threadIdx.x;
    if (g >= G) return;
    float s = fcb[0];
    for (int c = 0; c < D_HID; ++c) s += funflip(pooled[g * D_HID + c]) * fcw[c];
    out[g] = 1.f / (1.f + expf(-s));
}

// ---------------------------------------------------------------------------
// Host orchestration
// ---------------------------------------------------------------------------
static void gat_layer(const float* xin, int Kdim, const float* W,
                      const float* aS, const float* aD, const float* bias,
                      const int* srcIdx, const int* dstIdx, int Nn, int Etot,
                      float* hbuf, float* hout, float* agg, float* evals,
                      float* als, float* ald, unsigned* mflip, float* denom,
                      float* colsum, float* colsq, hipStream_t stream)
{
    const int tilesM = (Nn + 15) / 16;
    const int waves  = tilesM * (D_HID / 64);   // one wave per 16x64 strip
    gemm_wmma_f32<<<(waves + 7) / 8, 256, 0, stream>>>(xin, W, hbuf, Nn, Kdim, D_HID);

    attn_coef<<<(Nn * HEADS + 255) / 256, 256, 0, stream>>>(hbuf, aS, aD, als, ald, Nn);

    const int aggN = Nn * D_HID;
    init_layer<<<(aggN + 255) / 256, 256, 0, stream>>>(agg, mflip, denom, colsum, colsq, aggN, Nn);

    edge_logits<<<(Etot * HEADS + 255) / 256, 256, 0, stream>>>(srcIdx, dstIdx, als, ald, evals, mflip, Etot);
    edge_exp   <<<(Etot * HEADS + 255) / 256, 256, 0, stream>>>(dstIdx, evals, mflip, denom, Etot);

    const long long aggThreads = (long long)Etot * HEADS * 32;
    edge_aggregate<<<(int)((aggThreads + 255) / 256), 256, 0, stream>>>(srcIdx, dstIdx, hbuf, evals, denom, agg, Etot);

    post_bias_relu<<<(aggN + 255) / 256, 256, 0, stream>>>(agg, bias, hout, aggN);
}

extern "C" void kernel_launch(void* const* d_in, const int* in_sizes, int n_in,
                              void* d_out, int out_size, void* d_ws, size_t ws_size,
                              hipStream_t stream)
{
    const float* x     = (const float*)d_in[0];
    const int*   ei    = (const int*)  d_in[1];
    const int*   batch = (const int*)  d_in[2];
    const float* W0  = (const float*)d_in[3];
    const float* as0 = (const float*)d_in[4];
    const float* ad0 = (const float*)d_in[5];
    const float* b0  = (const float*)d_in[6];
    const float* W1  = (const float*)d_in[7];
    const float* as1 = (const float*)d_in[8];
    const float* ad1 = (const float*)d_in[9];
    const float* b1  = (const float*)d_in[10];
    const float* W2  = (const float*)d_in[11];
    const float* as2 = (const float*)d_in[12];
    const float* ad2 = (const float*)d_in[13];
    const float* b2  = (const float*)d_in[14];
    const float* g0  = (const float*)d_in[15];
    const float* be0 = (const float*)d_in[16];
    const float* g1  = (const float*)d_in[17];
    const float* be1 = (const float*)d_in[18];
    const float* fcw = (const float*)d_in[19];
    const float* fcb = (const float*)d_in[20];

    const int Nn   = in_sizes[2];           // 20000 nodes
    const int Etot = in_sizes[1] / 2;       // 660000 edges incl. self loops
    const int Fin  = in_sizes[0] / Nn;      // 256
    const int G    = out_size;              // 64 graphs
    const int* srcIdx = ei;
    const int* dstIdx = ei + Etot;

    // workspace carve-up (all fp32/u32; ~73 MB total)
    float* buf0  = (float*)d_ws;                       // [N,256] GEMM output h
    float* buf1  = buf0 + (size_t)Nn * D_HID;          // [N,256] layer output / next input
    float* agg   = buf1 + (size_t)Nn * D_HID;          // [N,256] message accumulation
    float* evals = agg  + (size_t)Nn * D_HID;          // [Etot,4] edge logits -> exp
    float* als   = evals + (size_t)Etot * HEADS;       // [N,4]
    float* ald   = als + (size_t)Nn * HEADS;           // [N,4]
    unsigned* mflip = (unsigned*)(ald + (size_t)Nn * HEADS);  // [N,4] seg-max (flipped)
    float* denom = (float*)(mflip + (size_t)Nn * HEADS);      // [N,4]
    float* colsum = denom + (size_t)Nn * HEADS;        // [256]
    float* colsq  = colsum + D_HID;                    // [256]
    unsigned* pooled = (unsigned*)(colsq + D_HID);     // [G,256] (flipped)

    const int aggN = Nn * D_HID;

    // ---- layer 0: x -> buf1 (relu) -> BN in-place
    gat_layer(x, Fin, W0, as0, ad0, b0, srcIdx, dstIdx, Nn, Etot,
              buf0, buf1, agg, evals, als, ald, mflip, denom, colsum, colsq, stream);
    col_stats<<<128, 256, 0, stream>>>(buf1, colsum, colsq, Nn);
    bn_apply <<<(aggN + 255) / 256, 256, 0, stream>>>(buf1, colsum, colsq, g0, be0, Nn);

    // ---- layer 1: buf1 -> buf1 (relu) -> BN in-place
    gat_layer(buf1, D_HID, W1, as1, ad1, b1, srcIdx, dstIdx, Nn, Etot,
              buf0, buf1, agg, evals, als, ald, mflip, denom, colsum, colsq, stream);
    col_stats<<<128, 256, 0, stream>>>(buf1, colsum, colsq, Nn);
    bn_apply <<<(aggN + 255) / 256, 256, 0, stream>>>(buf1, colsum, colsq, g1, be1, Nn);

    // ---- layer 2: buf1 -> buf1 (relu), then graph max-pool + FC + sigmoid
    gat_layer(buf1, D_HID, W2, as2, ad2, b2, srcIdx, dstIdx, Nn, Etot,
              buf0, buf1, agg, evals, als, ald, mflip, denom, colsum, colsq, stream);

    pool_init<<<(G * D_HID + 255) / 256, 256, 0, stream>>>(pooled, G * D_HID);
    pool_max <<<(aggN + 255) / 256, 256, 0, stream>>>(buf1, batch, pooled, Nn);
    fc_sigmoid<<<(G + 63) / 64, 64, 0, stream>>>(pooled, fcw, fcb, (float*)d_out, G);
}